// DeformableConv2d_65137474011680
// MI455X (gfx1250) — compile-verified
//
#include <hip/hip_runtime.h>

typedef __attribute__((ext_vector_type(2))) float v2f;
typedef __attribute__((ext_vector_type(8))) float v8f;

#define BATCH 8
#define CIN   64
#define HH    96
#define WW    96
#define COUT  64
#define NPIX  (HH*WW)      // 9216
#define KDIM  (CIN*9)      // 576
#define NTILE 16
#define NCHUNK (KDIM/4)    // 144 wmma_f32_16x16x4_f32 steps per tile

__launch_bounds__(128)
__global__ void deform_conv_fused(const float* __restrict__ x,
                                  const float* __restrict__ weight,
                                  const float* __restrict__ bias,
                                  const float* __restrict__ off_w,
                                  const float* __restrict__ off_b,
                                  float* __restrict__ out)
{
    __shared__ float sB[KDIM * NTILE];   // sampled B-tile: [k = c*9+kk][n], 36 KB
    __shared__ int   sIY[144];
    __shared__ int   sIX[144];
    __shared__ float sWt[4][144];        // bilinear corner weights

    const int tid  = threadIdx.x;
    const int b    = blockIdx.x / (NPIX / NTILE);
    const int pos0 = (blockIdx.x % (NPIX / NTILE)) * NTILE;

    const float* xb = x + (size_t)b * CIN * NPIX;

    // ---- Phase A/B: offset conv (2 channels per tap) + bilinear coord setup ----
    for (int s = tid; s < 144; s += 128) {
        const int kk  = s >> 4;          // tap index 0..8
        const int n   = s & 15;          // position within tile
        const int pos = pos0 + n;
        const int h   = pos / WW;
        const int w   = pos % WW;
        const int ky  = kk / 3 - 1;
        const int kx  = kk % 3 - 1;
        const int ry  = kk * 2;          // y-offset channel
        const int rx  = kk * 2 + 1;      // x-offset channel

        float offY = off_b[ry];
        float offX = off_b[rx];
        for (int c = 0; c < CIN; ++c) {
            const float* xc = xb + (size_t)c * NPIX;
            const float* wy = off_w + ((size_t)ry * CIN + c) * 9;
            const float* wx = off_w + ((size_t)rx * CIN + c) * 9;
            #pragma unroll
            for (int dy = -1; dy <= 1; ++dy) {
                const int yy = h + dy;
                if (yy < 0 || yy >= HH) continue;
                #pragma unroll
                for (int dx = -1; dx <= 1; ++dx) {
                    const int xx = w + dx;
                    if (xx < 0 || xx >= WW) continue;
                    const float xin = xc[yy * WW + xx];
                    const int   t   = (dy + 1) * 3 + (dx + 1);
                    offY = fmaf(xin, wy[t], offY);
                    offX = fmaf(xin, wx[t], offX);
                }
            }
        }

        const float sy = (float)(h + 1 + ky) + offY;   // by = h*STRIDE + (K/2)*DIL
        const float sx = (float)(w + 1 + kx) + offX;
        const float fy = floorf(sy), fx = floorf(sx);
        const int   iy0 = (int)fy,  ix0 = (int)fx;
        const float wy1 = sy - fy,  wx1 = sx - fx;
        const float wy0 = 1.0f - wy1, wx0 = 1.0f - wx1;
        sIY[s] = iy0;  sIX[s] = ix0;
        sWt[0][s] = wy0 * wx0;
        sWt[1][s] = wy0 * wx1;
        sWt[2][s] = wy1 * wx0;
        sWt[3][s] = wy1 * wx1;
    }
    __syncthreads();

    // ---- Phase C: bilinear gather of all channels into LDS B-tile ----
    for (int i = tid; i < CIN * 144; i += 128) {
        const int c  = i / 144;
        const int s  = i % 144;
        const int kk = s >> 4;
        const int n  = s & 15;
        const int iy0 = sIY[s], ix0 = sIX[s];
        const float* img = xb + (size_t)c * NPIX;

        float v = 0.0f;
        const bool x0v = (ix0 >= 0)  && (ix0 < WW);
        const bool x1v = (ix0 >= -1) && (ix0 < WW - 1);
        if (iy0 >= 0 && iy0 < HH) {
            const float* row = img + iy0 * WW;
            if (x0v) v = fmaf(sWt[0][s], row[ix0],     v);
            if (x1v) v = fmaf(sWt[1][s], row[ix0 + 1], v);
        }
        if (iy0 >= -1 && iy0 < HH - 1) {
            const float* row = img + (iy0 + 1) * WW;
            if (x0v) v = fmaf(sWt[2][s], row[ix0],     v);
            if (x1v) v = fmaf(sWt[3][s], row[ix0 + 1], v);
        }
        sB[(c * 9 + kk) * NTILE + n] = v;
    }
    __syncthreads();

    // ---- Phase D: fp32 WMMA GEMM, one 16x16 D-tile per wave ----
    const int wave  = tid >> 5;
    const int lane  = tid & 31;
    const int m0    = wave * 16;            // output-channel block
    const int mn    = lane & 15;            // M index for A, N index for B
    const int halfk = (lane >> 4) << 1;     // lanes 0-15 -> K{0,1}; 16-31 -> K{2,3}
    const float* wrow = weight + (size_t)(m0 + mn) * KDIM;   // wflat[m][k]

    v8f acc = {0.f, 0.f, 0.f, 0.f, 0.f, 0.f, 0.f, 0.f};
    for (int kc = 0; kc < NCHUNK; ++kc) {
        const int k0 = kc * 4 + halfk;
        v2f a = *(const v2f*)(wrow + k0);    // contiguous pair (k0, k0+1)
        v2f bb;
        bb.x = sB[k0 * NTILE + mn];
        bb.y = sB[(k0 + 1) * NTILE + mn];
        acc = __builtin_amdgcn_wmma_f32_16x16x4_f32(
                  /*neg_a=*/false, a, /*neg_b=*/false, bb,
                  /*c_mod=*/(short)0, acc, /*reuse_a=*/false, /*reuse_b=*/false);
    }

    // D layout: VGPR j -> M = m0 + j (lanes 0-15) or m0 + j + 8 (lanes 16-31), N = lane&15
    float* outb = out + (size_t)b * COUT * NPIX + pos0;
    #pragma unroll
    for (int j = 0; j < 8; ++j) {
        const int m = m0 + j + ((lane >> 4) << 3);
        outb[(size_t)m * NPIX + mn] = acc[j] + bias[m];
    }
}

extern "C" void kernel_launch(void* const* d_in, const int* in_sizes, int n_in,
                              void* d_out, int out_size, void* d_ws, size_t ws_size,
                              hipStream_t stream) {
    (void)in_sizes; (void)n_in; (void)out_size; (void)d_ws; (void)ws_size;
    const float* x      = (const float*)d_in[0];
    const float* weight = (const float*)d_in[1];
    const float* bias   = (const float*)d_in[2];
    const float* off_w  = (const float*)d_in[3];
    const float* off_b  = (const float*)d_in[4];
    float* out = (float*)d_out;

    const int nblocks = BATCH * (NPIX / NTILE);   // 4608 tiles
    deform_conv_fused<<<nblocks, 128, 0, stream>>>(x, weight, bias, off_w, off_b, out);
}